// DynamicContrastiveLoss_38946763440455
// MI455X (gfx1250) — compile-verified
//
#include <hip/hip_runtime.h>

typedef __attribute__((ext_vector_type(2))) float v2f;
typedef __attribute__((ext_vector_type(8))) float v8f;

#define B_ROWS 262144
#define DIM    128
#define KCLS   1000
#define KPAD   1024

// Workspace layout (floats):
//   [0, 131072)        sums -> overwritten in-place with centroids (KPAD x DIM)
//   [131072, 132096)   counts (KPAD)
//   [132096, 133120)   sq = ||cent_k||^2 (KPAD)
//   byte offset 532480: double acc[4] = {sumsq, snorm, simsum, kpresent}
#define OFF_COUNTS 131072
#define OFF_SQ     132096
#define OFF_ACC_F  133120
#define TOTAL_F    133128   // 133120 floats + 4 doubles (8 float slots)

__global__ void dcl_zero_ws(float* __restrict__ ws) {
    int i = blockIdx.x * blockDim.x + threadIdx.x;
    if (i < TOTAL_F) ws[i] = 0.0f;
}

// One pass over z: per-class sums (atomic f32 scatter, L2-resident),
// per-class counts, and total sum of squares (wave reduce + f64 atomic).
__global__ void dcl_accum(const float* __restrict__ z, const int* __restrict__ y,
                          float* __restrict__ sums, float* __restrict__ counts,
                          double* __restrict__ acc) {
    const int lane   = threadIdx.x & 31;
    const int wave   = (blockIdx.x * blockDim.x + threadIdx.x) >> 5;
    const int nwaves = (gridDim.x * blockDim.x) >> 5;
    float ssq = 0.0f;
    for (int row = wave; row < B_ROWS; row += nwaves) {
        const float4 v = *(const float4*)(z + (size_t)row * DIM + lane * 4);
        const int k = y[row];
        float* dst = sums + (size_t)k * DIM + lane * 4;
        atomicAdd(dst + 0, v.x);
        atomicAdd(dst + 1, v.y);
        atomicAdd(dst + 2, v.z);
        atomicAdd(dst + 3, v.w);
        ssq += v.x * v.x + v.y * v.y + v.z * v.z + v.w * v.w;
        if (lane == 0) atomicAdd(counts + k, 1.0f);
    }
    for (int off = 16; off; off >>= 1) ssq += __shfl_xor(ssq, off, 32);
    if (lane == 0) atomicAdd(&acc[0], (double)ssq);
}

// One wave per class: cent = sums/max(n,1) in place, sq[k]=||cent||^2,
// snorm += n*||cent||^2, kpresent += (n>0).
__global__ void dcl_centroid(float* __restrict__ cent, const float* __restrict__ counts,
                             float* __restrict__ sq, double* __restrict__ acc) {
    const int lane = threadIdx.x & 31;
    const int k    = (blockIdx.x * blockDim.x + threadIdx.x) >> 5;
    if (k >= KPAD) return;
    const float n   = counts[k];
    const float inv = 1.0f / fmaxf(n, 1.0f);
    float4 v = *(const float4*)(cent + (size_t)k * DIM + lane * 4);
    v.x *= inv; v.y *= inv; v.z *= inv; v.w *= inv;
    *(float4*)(cent + (size_t)k * DIM + lane * 4) = v;
    float s = v.x * v.x + v.y * v.y + v.z * v.z + v.w * v.w;
    for (int off = 16; off; off >>= 1) s += __shfl_xor(s, off, 32);
    if (lane == 0) {
        sq[k] = s;
        if (n > 0.0f) {
            atomicAdd(&acc[1], (double)(s * n)); // snorm = sum_k n_k ||c_k||^2
            atomicAdd(&acc[3], 1.0);             // number of present classes
        }
    }
}

// One wave per 16x16 tile of G = C C^T (only tj >= ti), K-loop of 32
// v_wmma_f32_16x16x4_f32 steps, then fused RBF + masked pair reduction
// straight out of the accumulator registers.
__global__ void dcl_gram_wmma(const float* __restrict__ cent, const float* __restrict__ counts,
                              const float* __restrict__ sq, double* __restrict__ acc) {
    const int lane = threadIdx.x & 31;
    const int wave = (blockIdx.x * blockDim.x + threadIdx.x) >> 5;
    const int ti = wave >> 6;  // 0..63
    const int tj = wave & 63;  // 0..63
    if (ti >= 64 || tj < ti) return;  // wave-uniform: EXEC stays all-ones for WMMA

    const int ibase = ti * 16, jbase = tj * 16;
    const int row_a = ibase + (lane & 15);
    const int row_b = jbase + (lane & 15);
    const int koff  = (lane >> 4) * 2;  // lanes 16-31 hold K=2,3 of each 4-chunk

    const float* pa = cent + (size_t)row_a * DIM + koff;
    const float* pb = cent + (size_t)row_b * DIM + koff;

    v8f c = {};
    #pragma unroll
    for (int kc = 0; kc < DIM; kc += 4) {
        v2f a = *(const v2f*)(pa + kc);
        v2f b = *(const v2f*)(pb + kc);
        c = __builtin_amdgcn_wmma_f32_16x16x4_f32(
                /*neg_a=*/false, a, /*neg_b=*/false, b,
                /*c_mod=*/(short)0, c, /*reuse_a=*/false, /*reuse_b=*/false);
    }

    // C/D layout: lane holds G[ibase + r + 8*(lane>=16)][jbase + (lane&15)] in c[r]
    const int col = lane & 15;
    const int hi  = lane >> 4;
    const int gj  = jbase + col;
    const bool pj = (gj < KCLS) && (counts[gj] > 0.0f);
    const float sqj = pj ? sq[gj] : 0.0f;
    float local = 0.0f;
    #pragma unroll
    for (int r = 0; r < 8; ++r) {
        const int gi = ibase + r + hi * 8;
        if (pj && gi < gj && gi < KCLS && counts[gi] > 0.0f) {
            const float d2 = fmaxf(sq[gi] + sqj - 2.0f * c[r], 0.0f);
            local += __expf(-0.5f * d2);  // sigma = 1
        }
    }
    for (int off = 16; off; off >>= 1) local += __shfl_xor(local, off, 32);
    if (lane == 0 && local != 0.0f) atomicAdd(&acc[2], (double)local);
}

__global__ void dcl_finalize(const double* __restrict__ acc, float* __restrict__ out) {
    if (threadIdx.x != 0 || blockIdx.x != 0) return;
    const double sumsq = acc[0], snorm = acc[1], simsum = acc[2], kf = acc[3];
    const double l_intra = (sumsq - snorm) / (double)B_ROWS;
    const double npairs  = kf * (kf - 1.0) * 0.5;
    const double l_inter = simsum / fmax(npairs, 1.0);
    const double denom = l_intra + l_inter + 1e-12;
    double a = fmin(fmax(l_intra / denom, 0.05), 0.95);
    double b = fmin(fmax(l_inter / denom, 0.05), 0.95);
    const double s = a + b + 1e-12;
    a /= s; b /= s;
    const double ae = 0.9 * 0.5 + 0.1 * a;
    const double be = 0.9 * 0.5 + 0.1 * b;
    const double s2 = ae + be + 1e-12;
    const double alpha = ae / s2, beta = be / s2;
    out[0] = (float)(alpha * l_intra + beta * l_inter);
}

extern "C" void kernel_launch(void* const* d_in, const int* in_sizes, int n_in,
                              void* d_out, int out_size, void* d_ws, size_t ws_size,
                              hipStream_t stream) {
    const float* z = (const float*)d_in[0];
    const int*   y = (const int*)d_in[1];
    float* out = (float*)d_out;

    float*  ws     = (float*)d_ws;
    float*  sums   = ws;                       // becomes centroids in-place
    float*  counts = ws + OFF_COUNTS;
    float*  sq     = ws + OFF_SQ;
    double* acc    = (double*)(ws + OFF_ACC_F);

    // 1) zero accumulators (d_ws is poisoned, must re-zero every launch)
    dcl_zero_ws<<<(TOTAL_F + 255) / 256, 256, 0, stream>>>(ws);

    // 2) single streaming pass over z (134 MB): sums/counts/sum||z||^2
    dcl_accum<<<1024, 256, 0, stream>>>(z, y, sums, counts, acc);

    // 3) normalize to centroids, per-class norms, present count
    dcl_centroid<<<(KPAD * 32 + 255) / 256, 256, 0, stream>>>(sums, counts, sq, acc);

    // 4) WMMA Gram (upper-triangular tiles) + fused RBF pair sum
    //    64x64 tile grid, 1 wave per tile, 4 waves per block
    dcl_gram_wmma<<<(64 * 64) / 4, 128, 0, stream>>>(sums, counts, sq, acc);

    // 5) scalar EMA/clamp math -> loss
    dcl_finalize<<<1, 1, 0, stream>>>(acc, out);
}